// StructureExtractor_75187697484196
// MI455X (gfx1250) — compile-verified
//
#include <hip/hip_runtime.h>
#include <hip/hip_bf16.h>

#define NN 50000
#define DD 128
#define EE 800000
#define LLAYERS 3
#define EPSV 1e-5f
#define SLOPE 0.2f

typedef __attribute__((ext_vector_type(2))) float v2f;
typedef __attribute__((ext_vector_type(8))) float v8f;

// ---------- helpers ----------
__device__ __forceinline__ unsigned fkey(float f) {
  unsigned u = __float_as_uint(f);
  return (u & 0x80000000u) ? ~u : (u | 0x80000000u);   // order-preserving float->uint
}
__device__ __forceinline__ float funkey(unsigned u) {
  return (u & 0x80000000u) ? __uint_as_float(u ^ 0x80000000u) : __uint_as_float(~u);
}
__device__ __forceinline__ float lrelu(float v) { return v > 0.f ? v : SLOPE * v; }

// ---------- elementwise / init kernels ----------
__global__ void fill_f32(float* p, float v, int n) {
  int i = blockIdx.x * blockDim.x + threadIdx.x;
  if (i < n) p[i] = v;
}
__global__ void fill_u32(unsigned* p, unsigned v, int n) {
  int i = blockIdx.x * blockDim.x + threadIdx.x;
  if (i < n) p[i] = v;
}
__global__ void copy_f32(const float* __restrict__ s, float* __restrict__ d, int n) {
  int i = blockIdx.x * blockDim.x + threadIdx.x;
  if (i < n) d[i] = s[i];
}
__global__ void bcast_row(float* __restrict__ out, const float* __restrict__ vec, int n) {
  int i = blockIdx.x * blockDim.x + threadIdx.x;
  if (i < n) out[i] = vec[i & (DD - 1)];
}
__global__ void add_row(float* __restrict__ out, const float* __restrict__ vec, int n) {
  int i = blockIdx.x * blockDim.x + threadIdx.x;
  if (i < n) out[i] += vec[i & (DD - 1)];
}
__global__ void count_deg(const int* __restrict__ dst, float* __restrict__ deg) {
  int e = blockIdx.x * blockDim.x + threadIdx.x;
  if (e < EE) atomicAdd(&deg[dst[e]], 1.0f);
}
__global__ void rsqrt_ip(float* p, int n) {
  int i = blockIdx.x * blockDim.x + threadIdx.x;
  if (i < n) p[i] = rsqrtf(p[i]);
}

// ---------- fp32 WMMA GEMM: C[nrows x 128] = (relu?)A[nrows x 128] * B[128 x 128] ----------
// Block = one 16-row M tile x all 8 N tiles (8 waves). B (64 KB) staged in LDS once per
// block with coalesced b128 stores; inner loop = 2x ds_load_b32 (B frag) + 1x 8B A load
// + 1x V_WMMA_F32_16X16X4_F32 per k-step, 32 steps over K=128. ReLU is compile-time.
template <int RELU>
__global__ __launch_bounds__(256)
void gemm_wmma_f32(const float* __restrict__ A, const float* __restrict__ B,
                   float* __restrict__ C, int nrows) {
  __shared__ float Bs[DD * DD];                // 64 KB, row-major [k][n]
  const int tid = threadIdx.x;

  // cooperative coalesced load of B: 16384 floats = 4096 float4, 16 per thread
#pragma unroll
  for (int i = 0; i < (DD * DD) / (256 * 4); ++i) {
    int idx = (i * 256 + tid) * 4;
    *(float4*)(Bs + idx) = *(const float4*)(B + idx);
  }
  __syncthreads();

  const int wv   = tid >> 5;                   // 0..7 -> tile_n
  const int lane = tid & 31;
  const int m0 = blockIdx.x * 16;
  const int n0 = wv * 16;
  if (m0 >= nrows) return;                     // wave-uniform guard
  const int r16 = lane & 15;                   // A row / B,C column within tile
  const int kh  = lane >> 4;                   // K half-select (0: K{0,1}, 1: K{2,3})

  v8f acc = {};
  const float* arow = A + (size_t)(m0 + r16) * DD + 2 * kh;
  const float* bcol = Bs + (size_t)(2 * kh) * DD + n0 + r16;
#pragma unroll
  for (int k = 0; k < DD; k += 4) {
    v2f a = *(const v2f*)(arow + k);           // A[m][k+2kh], A[m][k+2kh+1]
    if (RELU) { a.x = fmaxf(a.x, 0.f); a.y = fmaxf(a.y, 0.f); }
    v2f b;
    b.x = bcol[k * DD];                        // B[k+2kh][n]
    b.y = bcol[k * DD + DD];                   // B[k+2kh+1][n]
    acc = __builtin_amdgcn_wmma_f32_16x16x4_f32(false, a, false, b, (short)0, acc,
                                                false, false);
  }
  float* ccol = C + (size_t)m0 * DD + n0 + r16;
#pragma unroll
  for (int v = 0; v < 8; ++v)
    ccol[(size_t)(v + 8 * kh) * DD] = acc[v];
}

// ---------- GCN edge scatter: agg[dst] += dinv[src]*dinv[dst]*h[src] ----------
// One wave per edge, 32 lanes x float4 = 128 features (rows are L2-resident).
__global__ __launch_bounds__(256)
void gcn_scatter(const int* __restrict__ ei, const float* __restrict__ h,
                 const float* __restrict__ dinv, float* __restrict__ agg) {
  int wave = (blockIdx.x * blockDim.x + threadIdx.x) >> 5;
  int lane = threadIdx.x & 31;
  if (wave >= EE + NN) return;
  int s, d;
  if (wave < EE) { s = ei[wave]; d = ei[EE + wave]; } else { s = d = wave - EE; }
  float nrm = dinv[s] * dinv[d];
  float4 v = ((const float4*)(h + (size_t)s * DD))[lane];
  float* o = agg + (size_t)d * DD + lane * 4;
  atomicAdd(o + 0, nrm * v.x);
  atomicAdd(o + 1, nrm * v.y);
  atomicAdd(o + 2, nrm * v.z);
  atomicAdd(o + 3, nrm * v.w);
}

// ---------- GATv2 edge logits + segment max ----------
__global__ __launch_bounds__(256)
void gat_logit(const int* __restrict__ ei, const float* __restrict__ xl,
               const float* __restrict__ xr, const float* __restrict__ att,
               float* __restrict__ elog, unsigned* __restrict__ mkey) {
  int wave = (blockIdx.x * blockDim.x + threadIdx.x) >> 5;
  int lane = threadIdx.x & 31;
  if (wave >= EE + NN) return;
  int s, d;
  if (wave < EE) { s = ei[wave]; d = ei[EE + wave]; } else { s = d = wave - EE; }
  float4 a = ((const float4*)att)[lane];
  float4 l = ((const float4*)(xl + (size_t)s * DD))[lane];
  float4 r = ((const float4*)(xr + (size_t)d * DD))[lane];
  float p = lrelu(l.x + r.x) * a.x + lrelu(l.y + r.y) * a.y +
            lrelu(l.z + r.z) * a.z + lrelu(l.w + r.w) * a.w;
#pragma unroll
  for (int off = 16; off; off >>= 1) p += __shfl_xor(p, off, 32);
  if (lane == 0) {
    elog[wave] = p;
    atomicMax(&mkey[d], fkey(p));
  }
}

// ---------- exp(e - m[dst]) and segment denominator ----------
__global__ void gat_exp(const int* __restrict__ ei, const float* __restrict__ elog,
                        const unsigned* __restrict__ mkey, float* __restrict__ exf,
                        float* __restrict__ den) {
  int e = blockIdx.x * blockDim.x + threadIdx.x;
  if (e >= EE + NN) return;
  int d = (e < EE) ? ei[EE + e] : e - EE;
  float m = funkey(mkey[d]);
  float x = expf(elog[e] - m);
  exf[e] = x;
  atomicAdd(&den[d], x);
}

// ---------- weighted scatter: xstruct[dst] += alpha * xl[src] ----------
__global__ __launch_bounds__(256)
void gat_scatter(const int* __restrict__ ei, const float* __restrict__ xl,
                 const float* __restrict__ exf, const float* __restrict__ den,
                 float* __restrict__ xs) {
  int wave = (blockIdx.x * blockDim.x + threadIdx.x) >> 5;
  int lane = threadIdx.x & 31;
  if (wave >= EE + NN) return;
  int s, d;
  if (wave < EE) { s = ei[wave]; d = ei[EE + wave]; } else { s = d = wave - EE; }
  float alpha = exf[wave] / den[d];
  float4 v = ((const float4*)(xl + (size_t)s * DD))[lane];
  float* o = xs + (size_t)d * DD + lane * 4;
  atomicAdd(o + 0, alpha * v.x);
  atomicAdd(o + 1, alpha * v.y);
  atomicAdd(o + 2, alpha * v.z);
  atomicAdd(o + 3, alpha * v.w);
}

// ---------- BatchNorm: per-column sum / sum-of-squares, then apply ----------
__global__ __launch_bounds__(256)
void bn_reduce(const float* __restrict__ xs, float* __restrict__ colsum,
               float* __restrict__ colsq) {
  int c = blockIdx.x;                           // 0..127
  int tid = threadIdx.x;
  float s = 0.f, q = 0.f;
  for (int i = tid; i < NN; i += blockDim.x) {
    float v = xs[(size_t)i * DD + c];
    s += v; q += v * v;
  }
  __shared__ float ss[256], sq[256];
  ss[tid] = s; sq[tid] = q; __syncthreads();
  for (int st = 128; st; st >>= 1) {
    if (tid < st) { ss[tid] += ss[tid + st]; sq[tid] += sq[tid + st]; }
    __syncthreads();
  }
  if (tid == 0) { colsum[c] = ss[0]; colsq[c] = sq[0]; }
}

__global__ void bn_apply(const float* __restrict__ xs, const float* __restrict__ colsum,
                         const float* __restrict__ colsq, const float* __restrict__ gamma,
                         const float* __restrict__ beta, float* __restrict__ out, int n) {
  int i = blockIdx.x * blockDim.x + threadIdx.x;
  if (i >= n) return;
  int c = i & (DD - 1);
  float mu = colsum[c] * (1.0f / NN);
  float var = colsq[c] * (1.0f / NN) - mu * mu;
  out[i] = (xs[i] - mu) * rsqrtf(var + EPSV) * gamma[c] + beta[c];
}

// ======================================================================
extern "C" void kernel_launch(void* const* d_in, const int* in_sizes, int n_in,
                              void* d_out, int out_size, void* d_ws, size_t ws_size,
                              hipStream_t stream) {
  (void)in_sizes; (void)n_in; (void)out_size; (void)ws_size;
  const float* x    = (const float*)d_in[0];
  const int*   ei   = (const int*)d_in[1];   // [2,E] flat: first E=src, next E=dst
  const float* Wg   = (const float*)d_in[2]; // [L,D,D]
  const float* bg   = (const float*)d_in[3]; // [L,D]
  const float* Wl   = (const float*)d_in[4];
  const float* Wr   = (const float*)d_in[5];
  const float* att  = (const float*)d_in[6];
  const float* bgat = (const float*)d_in[7];
  const float* gam  = (const float*)d_in[8];
  const float* bet  = (const float*)d_in[9];
  float* out = (float*)d_out;

  const size_t nd = (size_t)NN * DD;
  float* h    = (float*)d_ws;          // GCN h; reused as GAT xl
  float* xr   = h + nd;
  float* ra   = xr + nd;
  float* rb   = ra + nd;
  float* xs   = rb + nd;               // x_struct accumulator
  float* dinv = xs + nd;
  float* elog = dinv + NN;
  float* exf  = elog + (EE + NN);
  float* den  = exf + (EE + NN);
  unsigned* mkey = (unsigned*)(den + NN);
  float* colsum  = (float*)(mkey + NN);
  float* colsq   = colsum + DD;

  const int T = 256;
  const int ND   = NN * DD;
  const int bND  = (ND + T - 1) / T;                 // 25000
  const int bN   = (NN + T - 1) / T;
  const int bE   = (EE + T - 1) / T;
  const int bEN  = (EE + NN + T - 1) / T;            // thread-per-edge
  const int bEW  = ((EE + NN) * 32 + T - 1) / T;     // wave-per-edge
  const int bGEMM = NN / 16;                         // 3125 (one M tile per block)

  // degree -> dinv  (self-loop gives deg>=1)
  fill_f32<<<bN, T, 0, stream>>>(dinv, 1.0f, NN);
  count_deg<<<bE, T, 0, stream>>>(ei + EE, dinv);
  rsqrt_ip<<<bN, T, 0, stream>>>(dinv, NN);

  // x_struct = x
  copy_f32<<<bND, T, 0, stream>>>(x, xs, ND);

  float* rprev = ra;           // previous layer's r (relu applied at GEMM read)
  float* rcur  = rb;
  for (int l = 0; l < LLAYERS; ++l) {
    // h = (relu?)(input) @ Wg[l]
    if (l == 0)
      gemm_wmma_f32<0><<<bGEMM, T, 0, stream>>>(x, Wg, h, NN);
    else
      gemm_wmma_f32<1><<<bGEMM, T, 0, stream>>>(rprev, Wg + (size_t)l * DD * DD, h, NN);

    // r = bg[l]; r += norm-scatter(h)
    bcast_row<<<bND, T, 0, stream>>>(rcur, bg + (size_t)l * DD, ND);
    gcn_scatter<<<bEW, T, 0, stream>>>(ei, h, dinv, rcur);

    // xl = r @ Wl (into h), xr = r @ Wr
    gemm_wmma_f32<0><<<bGEMM, T, 0, stream>>>(rcur, Wl, h, NN);
    gemm_wmma_f32<0><<<bGEMM, T, 0, stream>>>(rcur, Wr, xr, NN);

    // segment softmax over dst
    fill_u32<<<bN, T, 0, stream>>>(mkey, 0u, NN);
    fill_f32<<<bN, T, 0, stream>>>(den, 0.0f, NN);
    gat_logit<<<bEW, T, 0, stream>>>(ei, h, xr, att, elog, mkey);
    gat_exp<<<bEN, T, 0, stream>>>(ei, elog, mkey, exf, den);

    // x_struct += segment_sum(alpha * xl[src]) + b_gat
    gat_scatter<<<bEW, T, 0, stream>>>(ei, h, exf, den, xs);
    add_row<<<bND, T, 0, stream>>>(xs, bgat, ND);

    float* t = rprev; rprev = rcur; rcur = t;
  }

  // BatchNorm (training-mode biased stats)
  bn_reduce<<<DD, T, 0, stream>>>(xs, colsum, colsq);
  bn_apply<<<bND, T, 0, stream>>>(xs, colsum, colsq, gam, bet, out, ND);
}